// OptimizedMoE_63685775065116
// MI455X (gfx1250) — compile-verified
//
#include <hip/hip_runtime.h>
#include <hip/hip_bf16.h>

// ---------------------------------------------------------------------------
// MoE forward, faithful to reference quirk (each expert gets ONE weighted-sum
// token vector).  E=16, D=2048, F=4096, T=8192.
// ---------------------------------------------------------------------------

#define E_EXPERTS 16
#define D_IN      2048
#define F_EXP     4096
#define T_TOK     8192   // B*S = 2*4096
#define K_SPLIT   8      // split-K chunks for dispatch@X

typedef float v2f __attribute__((ext_vector_type(2)));
typedef float v4f __attribute__((ext_vector_type(4)));
typedef float v8f __attribute__((ext_vector_type(8)));

// ---------------------------------------------------------------------------
// Kernel 0: zero the dense dispatch matrix [E][T] (scatter target)
// ---------------------------------------------------------------------------
__global__ void k_zero_dispatch(float* __restrict__ dispatch) {
    int i = blockIdx.x * blockDim.x + threadIdx.x;   // 512*256 == E*T
    dispatch[i] = 0.0f;
}

// ---------------------------------------------------------------------------
// Kernel 1: gating GEMM (WMMA f32 16x16x4) + analytic top-2 softmax renorm
//   One wave (32 lanes) per 16-token tile.  M=16 tokens, N=16 experts,
//   K loop over D=2048 in steps of 4.
//   softmax+top2+renorm collapses to: g0 = sigmoid(l_top1 - l_top2).
// ---------------------------------------------------------------------------
__global__ void k_gating_wmma(const float* __restrict__ x,        // [T][D]
                              const float* __restrict__ gate_w,   // [E][D]
                              const float* __restrict__ expert_weights, // [E]
                              float* __restrict__ dispatch,       // [E][T]
                              int*   __restrict__ top_i0,
                              int*   __restrict__ top_i1,
                              float* __restrict__ top_g0,
                              float* __restrict__ top_g1) {
    const int lane  = threadIdx.x;         // 0..31
    const int l15   = lane & 15;
    const int khalf = (lane >> 4) * 2;     // 0 for lanes 0-15, 2 for 16-31
    const int m0    = blockIdx.x * 16;     // token tile base

    // A operand rows: lane -> token row (M = l15), K pair = khalf{+1}
    const float* xrow = x + (size_t)(m0 + l15) * D_IN;
    // B operand cols: lane -> expert col (N = l15), same K pair
    const float* grow = gate_w + (size_t)l15 * D_IN;

    v8f c = {0.f, 0.f, 0.f, 0.f, 0.f, 0.f, 0.f, 0.f};

#pragma unroll 4
    for (int k = 0; k < D_IN; k += 4) {
        v2f a, b;
        a.x = xrow[k + khalf];
        a.y = xrow[k + khalf + 1];
        b.x = grow[k + khalf];
        b.y = grow[k + khalf + 1];
        // D = A(16x4) * B(4x16) + C  -> v_wmma_f32_16x16x4_f32
        c = __builtin_amdgcn_wmma_f32_16x16x4_f32(
                /*neg_a=*/false, a, /*neg_b=*/false, b,
                /*c_mod=*/(short)0, c, /*reuse_a=*/false, /*reuse_b=*/false);
    }

    // scale logits by learned expert weight (per-lane expert = l15)
    const float ew = expert_weights[l15];

    // C layout: lanes 0-15 hold rows M=r in c[r]; lanes 16-31 hold rows M=r+8.
    // For each of the 8 rows owned by this half-wave: top-2 over 16 lanes.
#pragma unroll
    for (int r = 0; r < 8; ++r) {
        float v = c[r] * ew;

        // --- argmax over the 16-lane half ---
        float v1 = v; int i1 = l15;
        for (int off = 8; off >= 1; off >>= 1) {
            float ov = __shfl_xor(v1, off, 16);
            int   oi = __shfl_xor(i1, off, 16);
            if (ov > v1 || (ov == v1 && oi < i1)) { v1 = ov; i1 = oi; }
        }
        // --- second max (mask out winner's lane) ---
        float vm = (l15 == i1) ? -INFINITY : v;
        float v2 = vm; int i2 = l15;
        for (int off = 8; off >= 1; off >>= 1) {
            float ov = __shfl_xor(v2, off, 16);
            int   oi = __shfl_xor(i2, off, 16);
            if (ov > v2 || (ov == v2 && oi < i2)) { v2 = ov; i2 = oi; }
        }

        // renormalized top-2 softmax: g0 = sigmoid(v1 - v2)
        float g0 = 1.0f / (1.0f + __expf(v2 - v1));
        float g1 = 1.0f - g0;

        if (l15 == 0) {
            int t = m0 + r + ((lane >> 4) * 8);
            top_i0[t] = i1;  top_i1[t] = i2;
            top_g0[t] = g0;  top_g1[t] = g1;
            dispatch[(size_t)i1 * T_TOK + t] = g0;   // distinct (e,t) pairs,
            dispatch[(size_t)i2 * T_TOK + t] = g1;   // no atomics needed
        }
    }
}

// ---------------------------------------------------------------------------
// Kernel 2a: split-K partials of expert_in = dispatch[E,T] @ X[T,D]
//   (WMMA f32 16x16x4).  Grid = (D/16) x K_SPLIT waves; each wave owns a
//   16-wide D tile and a T/K_SPLIT token chunk.  1024 waves -> enough
//   concurrency to cover HBM latency for the 64 MB X stream.
// ---------------------------------------------------------------------------
__global__ void k_expert_in_partial(const float* __restrict__ dispatch, // [E][T]
                                    const float* __restrict__ x,        // [T][D]
                                    float* __restrict__ part) {  // [K_SPLIT][E][D]
    const int lane  = threadIdx.x;
    const int l15   = lane & 15;
    const int khalf = (lane >> 4) * 2;
    const int n0    = (blockIdx.x % (D_IN / 16)) * 16;  // d-tile base
    const int chunk =  blockIdx.x / (D_IN / 16);        // 0..K_SPLIT-1
    const int kbeg  = chunk * (T_TOK / K_SPLIT);
    const int kend  = kbeg + (T_TOK / K_SPLIT);

    const float* drow = dispatch + (size_t)l15 * T_TOK;  // A: M = expert = l15
    const float* xcol = x + n0 + l15;                    // B: N = d col = l15

    v8f c = {0.f, 0.f, 0.f, 0.f, 0.f, 0.f, 0.f, 0.f};

#pragma unroll 4
    for (int k = kbeg; k < kend; k += 4) {
        v2f a, b;
        a.x = drow[k + khalf];
        a.y = drow[k + khalf + 1];
        b.x = xcol[(size_t)(k + khalf) * D_IN];
        b.y = xcol[(size_t)(k + khalf + 1) * D_IN];
        c = __builtin_amdgcn_wmma_f32_16x16x4_f32(
                false, a, false, b, (short)0, c, false, false);
    }

    // C layout write-back: lanes 0-15 -> rows r, lanes 16-31 -> rows r+8
    const int rbase = (lane >> 4) * 8;
    float* pbase = part + (size_t)chunk * (E_EXPERTS * D_IN);
#pragma unroll
    for (int r = 0; r < 8; ++r) {
        pbase[(size_t)(rbase + r) * D_IN + n0 + l15] = c[r];
    }
}

// ---------------------------------------------------------------------------
// Kernel 2b: reduce split-K partials -> expert_in [E][D]  (deterministic)
// ---------------------------------------------------------------------------
__global__ void k_expert_in_reduce(const float* __restrict__ part, // [K_SPLIT][E*D]
                                   float* __restrict__ expert_in) { // [E*D]
    int i = blockIdx.x * blockDim.x + threadIdx.x;   // E*D = 32768 threads
    float s = 0.0f;
#pragma unroll
    for (int cSplit = 0; cSplit < K_SPLIT; ++cSplit)
        s += part[(size_t)cSplit * (E_EXPERTS * D_IN) + i];
    expert_in[i] = s;
}

// ---------------------------------------------------------------------------
// Kernel 3: expert_out[e][f] = expert_in[e] . expert_w[e][f] + expert_b[e][f]
//   Streams the 512 MB expert_w (zero reuse) with NON-TEMPORAL b128 loads so
//   it never displaces the reused working set from L2.  One wave per f.
// ---------------------------------------------------------------------------
__global__ void k_expert_gemv(const float* __restrict__ expert_in, // [E][D]
                              const float* __restrict__ expert_w,  // [E][F][D]
                              const float* __restrict__ expert_b,  // [E][F]
                              float* __restrict__ expert_out) {    // [E][F]
    __shared__ float xin[D_IN];                       // 8 KB
    const int e     = blockIdx.x / (F_EXP / 8);
    const int fbase = (blockIdx.x % (F_EXP / 8)) * 8;
    const int tid   = threadIdx.x;

    for (int i = tid; i < D_IN; i += 256)
        xin[i] = expert_in[(size_t)e * D_IN + i];
    __syncthreads();

    const int wave = tid >> 5;
    const int lane = tid & 31;
    const int f    = fbase + wave;

    const v4f* wrow =
        (const v4f*)(expert_w + ((size_t)e * F_EXP + f) * D_IN);

    float acc = 0.0f;
#pragma unroll 4
    for (int d4 = lane; d4 < D_IN / 4; d4 += 32) {     // NT b128 stream
        v4f w = __builtin_nontemporal_load(&wrow[d4]);
        int d = d4 * 4;
        acc += w.x * xin[d] + w.y * xin[d + 1] + w.z * xin[d + 2] + w.w * xin[d + 3];
    }
    // wave32 reduction
    for (int off = 16; off >= 1; off >>= 1)
        acc += __shfl_xor(acc, off, 32);

    if (lane == 0)
        expert_out[(size_t)e * F_EXP + f] = acc + expert_b[(size_t)e * F_EXP + f];
}

// ---------------------------------------------------------------------------
// Kernel 4: combined[t][f] = g0*expert_out[i0][f] + g1*expert_out[i1][f]
//   expert_out (256 KB) stays cached (regular loads); the 128 MB output is a
//   one-shot stream -> non-temporal stores.
// ---------------------------------------------------------------------------
__global__ void k_combine(const float* __restrict__ expert_out, // [E][F]
                          const int*   __restrict__ top_i0,
                          const int*   __restrict__ top_i1,
                          const float* __restrict__ top_g0,
                          const float* __restrict__ top_g1,
                          float* __restrict__ out) {            // [T][F]
    const int t   = blockIdx.x;
    const int tid = threadIdx.x;

    const int   i0 = top_i0[t], i1 = top_i1[t];
    const float g0 = top_g0[t], g1 = top_g1[t];

    const v4f* eo0 = (const v4f*)(expert_out + (size_t)i0 * F_EXP);
    const v4f* eo1 = (const v4f*)(expert_out + (size_t)i1 * F_EXP);
    v4f*       o4  = (v4f*)(out + (size_t)t * F_EXP);

#pragma unroll
    for (int j = tid; j < F_EXP / 4; j += 256) {
        v4f a = eo0[j], b = eo1[j];
        v4f r = g0 * a + g1 * b;
        __builtin_nontemporal_store(r, &o4[j]);
    }
}

// ---------------------------------------------------------------------------
extern "C" void kernel_launch(void* const* d_in, const int* in_sizes, int n_in,
                              void* d_out, int out_size, void* d_ws, size_t ws_size,
                              hipStream_t stream) {
    const float* x              = (const float*)d_in[0];  // [B,S,D]
    const float* gate_w         = (const float*)d_in[1];  // [E,D]
    const float* expert_weights = (const float*)d_in[2];  // [E]
    const float* expert_w       = (const float*)d_in[3];  // [E,F,D]
    const float* expert_b       = (const float*)d_in[4];  // [E,F]
    float* out = (float*)d_out;

    // workspace carve-up (all 256B aligned)
    char* ws = (char*)d_ws;
    size_t off = 0;
    auto carve = [&](size_t bytes) { char* p = ws + off;
                                     off += (bytes + 255) & ~(size_t)255;
                                     return p; };
    float* dispatch   = (float*)carve((size_t)E_EXPERTS * T_TOK * 4);   // 512 KB
    int*   top_i0     = (int*)  carve((size_t)T_TOK * 4);
    int*   top_i1     = (int*)  carve((size_t)T_TOK * 4);
    float* top_g0     = (float*)carve((size_t)T_TOK * 4);
    float* top_g1     = (float*)carve((size_t)T_TOK * 4);
    float* ein_part   = (float*)carve((size_t)K_SPLIT * E_EXPERTS * D_IN * 4); // 1 MB
    float* expert_in  = (float*)carve((size_t)E_EXPERTS * D_IN * 4);    // 128 KB
    float* expert_out = (float*)carve((size_t)E_EXPERTS * F_EXP * 4);   // 256 KB
    (void)ws_size; (void)in_sizes; (void)n_in; (void)out_size;

    // 0) zero dispatch scatter target
    k_zero_dispatch<<<(E_EXPERTS * T_TOK) / 256, 256, 0, stream>>>(dispatch);

    // 1) gating GEMM (WMMA) + top-2 + scatter
    k_gating_wmma<<<T_TOK / 16, 32, 0, stream>>>(
        x, gate_w, expert_weights, dispatch, top_i0, top_i1, top_g0, top_g1);

    // 2) expert_in = dispatch @ X (WMMA, split-K) + reduce
    k_expert_in_partial<<<(D_IN / 16) * K_SPLIT, 32, 0, stream>>>(
        dispatch, x, ein_part);
    k_expert_in_reduce<<<(E_EXPERTS * D_IN) / 256, 256, 0, stream>>>(
        ein_part, expert_in);

    // 3) per-expert GEMV over 512 MB expert_w (bandwidth-dominant stage)
    k_expert_gemv<<<E_EXPERTS * (F_EXP / 8), 256, 0, stream>>>(
        expert_in, expert_w, expert_b, expert_out);

    // 4) combine back to tokens
    k_combine<<<T_TOK, 256, 0, stream>>>(
        expert_out, top_i0, top_i1, top_g0, top_g1, out);
}